// EdgeAwareNightConverter_40338332844403
// MI455X (gfx1250) — compile-verified
//
#include <hip/hip_runtime.h>
#include <hip/hip_bf16.h>

// ---------------------------------------------------------------------------
// Fused edge-aware night converter for gfx1250 (MI455X).
// One pass over the image; all 3x3 convolutions run as banded-matrix WMMAs
// (v_wmma_f32_16x16x4_f32): vertical 3-tap = A(16x4 const band) x B(4x16 data),
// horizontal 3-tap fused into B-fragment construction from LDS.
// ---------------------------------------------------------------------------

typedef __attribute__((ext_vector_type(2))) float v2f;
typedef __attribute__((ext_vector_type(8))) float v8f;

#define TILE   32
#define PITCH  52
#define ROWS   54
#define NTHR   256
#define NWAVE  8
#define IMG_HW (512 * 512)

__device__ __forceinline__ v8f wmma4(v2f a, v2f b, v8f c) {
  // D = A(16x4) * B(4x16) + C   (8 args: neg_a, A, neg_b, B, c_mod, C, reuse_a, reuse_b)
  return __builtin_amdgcn_wmma_f32_16x16x4_f32(false, a, false, b, (short)0, c,
                                               false, false);
}

// banded vertical weights (zero outside [0,2] handles K padding to 20 for free)
__device__ __forceinline__ float wsm(int d) {  // smooth [1,2,1]
  return d == 1 ? 2.f : (((d == 0) | (d == 2)) ? 1.f : 0.f);
}
__device__ __forceinline__ float wdf(int d) {  // diff [1,0,-1]
  return d == 0 ? 1.f : (d == 2 ? -1.f : 0.f);
}
__device__ __forceinline__ float wbx(int d) {  // box [1,1,1]/3
  return (d >= 0 && d <= 2) ? (1.f / 3.f) : 0.f;
}

// One masked box-blur pass over LDS: dst = hard_edge||out_of_image ? 0 : box3(src).
// Tiles of 16x16 starting at (lo,lo), nt tiles per dim, one tile per wave round.
__device__ __forceinline__ void box_pass(const float* __restrict__ src,
                                         float* __restrict__ dst,
                                         const float* __restrict__ Eb,
                                         int lo, int nt, int ty0, int tx0,
                                         int lane, int wv) {
  const int N = lane & 15;
  const int hf = lane >> 4;
  const int total = nt * nt;
  for (int p = wv; p < total; p += NWAVE) {
    const int r0 = lo + (p / nt) * 16;
    const int c0 = lo + (p % nt) * 16;
    v8f acc = {};
    const int col = c0 + N;
#pragma unroll
    for (int j = 0; j < 5; ++j) {
      const int k0 = 4 * j + 2 * hf;
      const int k1 = k0 + 1;
      const int row0 = (r0 - 1 + k0) * PITCH + col;
      const int row1 = (r0 - 1 + k1) * PITCH + col;
      v2f bb;
      bb.x = (src[row0 - 1] + src[row0] + src[row0 + 1]) * (1.f / 3.f);
      bb.y = (src[row1 - 1] + src[row1] + src[row1 + 1]) * (1.f / 3.f);
      v2f aa;
      aa.x = wbx(k0 - N);
      aa.y = wbx(k1 - N);
      acc = wmma4(aa, bb, acc);
    }
#pragma unroll
    for (int v = 0; v < 8; ++v) {
      const int i = r0 + v + 8 * hf;
      const int jj = c0 + N;
      const int y = ty0 + i - 3;
      const int x = tx0 + jj - 3;
      const bool in = ((unsigned)y < 512u) && ((unsigned)x < 512u);
      const float he = Eb[i * PITCH + jj];
      dst[i * PITCH + jj] = (in && he < 0.5f) ? acc[v] : 0.f;
    }
  }
}

__global__ __launch_bounds__(NTHR) void EdgeAwareNight_kernel(
    const float* __restrict__ img, const float* __restrict__ nip,
    float* __restrict__ out) {
  const int tx0 = blockIdx.x * TILE;
  const int ty0 = blockIdx.y * TILE;
  const int b = blockIdx.z;
  const int tid = threadIdx.x;
  const int lane = tid & 31;
  const int wv = tid >> 5;

  __shared__ float G[ROWS * PITCH];   // grayscale, halo 3
  __shared__ float E[ROWS * PITCH];   // hard-edge mask (1/0)
  __shared__ float DA[ROWS * PITCH];  // distance field ping
  __shared__ float DB[ROWS * PITCH];  // distance field pong

  // Zero-init: guarantees finite values under zero-weight / padded WMMA reads.
  for (int i = tid; i < ROWS * PITCH; i += NTHR) {
    G[i] = 0.f; E[i] = 0.f; DA[i] = 0.f; DB[i] = 0.f;
  }
  __syncthreads();

  // ---- Stage 1: grayscale tile (38x38, halo 3), zero outside the image ----
  const float* base = img + (size_t)b * 3 * IMG_HW;
  for (int e = tid; e < 38 * 38; e += NTHR) {
    const int i = e / 38, j = e % 38;
    const int y = ty0 + i - 3, x = tx0 + j - 3;
    float g = 0.f;
    if (((unsigned)y < 512u) && ((unsigned)x < 512u)) {
      const int off = y * 512 + x;
      g = (base[off] + base[IMG_HW + off] + base[2 * IMG_HW + off]) * (1.f / 3.f);
    }
    G[i * PITCH + j] = g;
  }
  __syncthreads();

  // ---- Stage 2: Sobel -> hard edge over halo-2 region [1,37), via WMMA ----
  {
    const int N = lane & 15;
    const int hf = lane >> 4;
    for (int p = wv; p < 9; p += NWAVE) {
      const int r0 = 1 + (p / 3) * 16;
      const int c0 = 1 + (p % 3) * 16;
      v8f axc = {};  // gx accumulator: vsmooth x hdiff
      v8f ayc = {};  // gy accumulator: vdiff   x hsmooth
      const int col = c0 + N;
#pragma unroll
      for (int j = 0; j < 5; ++j) {
        const int k0 = 4 * j + 2 * hf;
        const int k1 = k0 + 1;
        const int row0 = (r0 - 1 + k0) * PITCH + col;
        const int row1 = (r0 - 1 + k1) * PITCH + col;
        const float l0 = G[row0 - 1], c0v = G[row0], rr0 = G[row0 + 1];
        const float l1 = G[row1 - 1], c1v = G[row1], rr1 = G[row1 + 1];
        v2f bx, by, a_s, a_d;
        bx.x = l0 - rr0;                bx.y = l1 - rr1;
        by.x = l0 + 2.f * c0v + rr0;    by.y = l1 + 2.f * c1v + rr1;
        a_s.x = wsm(k0 - N);            a_s.y = wsm(k1 - N);
        a_d.x = wdf(k0 - N);            a_d.y = wdf(k1 - N);
        axc = wmma4(a_s, bx, axc);
        ayc = wmma4(a_d, by, ayc);
      }
#pragma unroll
      for (int v = 0; v < 8; ++v) {
        const int i = r0 + v + 8 * hf;
        const float gx = axc[v], gy = ayc[v];
        // edge_strength > 0.1  <=>  gx^2 + gy^2 > 0.01 (sign-free)
        E[i * PITCH + (c0 + N)] = (gx * gx + gy * gy > 0.01f) ? 1.f : 0.f;
      }
    }
  }
  __syncthreads();

  // ---- Stage 3: df1 = mask( box3(ones) ) is analytic: cnt_y*cnt_x/9 ----
  for (int e = tid; e < 36 * 36; e += NTHR) {
    const int i = 1 + e / 36, j = 1 + e % 36;
    const int y = ty0 + i - 3, x = tx0 + j - 3;
    const bool in = ((unsigned)y < 512u) && ((unsigned)x < 512u);
    const float he = E[i * PITCH + j];
    const float cy = 1.f + (y > 0) + (y < 511);
    const float cx = 1.f + (x > 0) + (x < 511);
    DA[i * PITCH + j] = (in && he < 0.5f) ? cy * cx * (1.f / 9.f) : 0.f;
  }
  __syncthreads();

  // ---- Stages 4/5: masked box blurs 2 and 3 via WMMA ----
  box_pass(DA, DB, E, 2, 3, ty0, tx0, lane, wv);  // valid over [2,36)
  __syncthreads();
  box_pass(DB, DA, E, 3, 2, ty0, tx0, lane, wv);  // valid over [3,35) == tile
  __syncthreads();

  // ---- Stage 6: final modulation, one read of img (hot in WGP$/L2) ----
  const float s = 1.f - (*nip) * 0.7f;
  float* ob = out + (size_t)b * 3 * IMG_HW;
  for (int e = tid; e < TILE * TILE; e += NTHR) {
    const int py = e >> 5, px = e & 31;
    const int y = ty0 + py, x = tx0 + px;
    const float df = DA[(py + 3) * PITCH + (px + 3)];
    const float dark = fminf(fmaxf(1.f - df * 0.8f, 0.3f), 1.0f);
    const float m = s * dark;
    const int off = y * 512 + x;
    ob[off] = base[off] * m;
    ob[IMG_HW + off] = base[IMG_HW + off] * m;
    ob[2 * IMG_HW + off] = base[2 * IMG_HW + off] * m;
  }
}

extern "C" void kernel_launch(void* const* d_in, const int* in_sizes, int n_in,
                              void* d_out, int out_size, void* d_ws, size_t ws_size,
                              hipStream_t stream) {
  (void)in_sizes; (void)n_in; (void)out_size; (void)d_ws; (void)ws_size;
  const float* img = (const float*)d_in[0];
  // d_in[1] = semantic_mask: unused by the reference computation.
  const float* ni = (const float*)d_in[2];
  float* out = (float*)d_out;
  dim3 grid(512 / TILE, 512 / TILE, 32);
  dim3 block(NTHR);
  EdgeAwareNight_kernel<<<grid, block, 0, stream>>>(img, ni, out);
}